// MyPVT18_57226144252767
// MI455X (gfx1250) — compile-verified
//
#include <hip/hip_runtime.h>
#include <cmath>

// Problem constants (match reference)
#define BB 16
#define NN 16384
#define CC 128
#define HH 128
#define WW 128
#define HWPIX (HH * WW)
#define EPSV 1e-6f

typedef __attribute__((ext_vector_type(2))) float v2f;
typedef __attribute__((ext_vector_type(4))) float v4f;
typedef __attribute__((ext_vector_type(8))) float v8f;

// 1-D Gaussian tap weight for offset d in {-1,0,1}, else 0.
__device__ __forceinline__ float wtap(int d, float k0, float k1) {
    d = d < 0 ? -d : d;
    return d == 0 ? k1 : (d == 1 ? k0 : 0.0f);
}

// ---------------------------------------------------------------------------
// Kernel 1: zero the accumulation map (B*HW*129 floats), float4 stores.
__global__ void k_zero(float* __restrict__ p) {
    size_t i = (size_t)blockIdx.x * blockDim.x + threadIdx.x;
    ((v4f*)p)[i] = (v4f){0.f, 0.f, 0.f, 0.f};
}

// ---------------------------------------------------------------------------
// Kernel 2: scatter tokens into map0 (b, hw, 129). 2 tokens per 256-block;
// thread c does one coalesced f32 atomic add, thread 0 also bumps the count.
__global__ void k_scatter(const float* __restrict__ x, const float* __restrict__ loc,
                          float* __restrict__ map0) {
    int token = blockIdx.x * 2 + (threadIdx.x >> 7);
    int c = threadIdx.x & 127;
    float lx = loc[(size_t)token * 2 + 0];
    float ly = loc[(size_t)token * 2 + 1];
    lx = fminf(fmaxf(lx, 0.f), 1.f);
    ly = fminf(fmaxf(ly, 0.f), 1.f);
    int xi = (int)rintf(lx * (WW - 1));   // RNE matches jnp.round
    int yi = (int)rintf(ly * (HH - 1));
    int b = token >> 14;  // token / NN
    size_t base = ((size_t)b * HWPIX + (size_t)(yi * WW + xi)) * 129;
    atomicAdd(&map0[base + c], x[(size_t)token * CC + c]);
    if (c == 0) atomicAdd(&map0[base + 128], 1.0f);
}

// ---------------------------------------------------------------------------
// Kernel 3a: normalize features by count, gate by count>0 (in place).
__global__ void k_norm_feat(float* __restrict__ map0) {
    size_t i = (size_t)blockIdx.x * blockDim.x + threadIdx.x;  // over B*HW*128
    size_t pix = i >> 7;
    int c = (int)(i & 127);
    float msum = map0[pix * 129 + 128];
    float bin = msum > 0.f ? 1.f : 0.f;
    size_t a = pix * 129 + c;
    map0[a] = map0[a] / (msum + EPSV) * bin;
}

// Kernel 3b: binarize the mask channel (separate kernel -> no RAW race).
__global__ void k_norm_mask(float* __restrict__ map0) {
    size_t pix = (size_t)blockIdx.x * blockDim.x + threadIdx.x;  // over B*HW
    size_t a = pix * 129 + 128;
    map0[a] = map0[a] > 0.f ? 1.f : 0.f;
}

// ---------------------------------------------------------------------------
// Kernel 4: 3x3 Gaussian blur of the single mask channel (scalar; 8 MB data).
__global__ void k_bmask(const float* __restrict__ map0, float* __restrict__ bmask,
                        float k0, float k1) {
    size_t i = (size_t)blockIdx.x * blockDim.x + threadIdx.x;  // over B*HW
    int b = (int)(i / HWPIX);
    int pix = (int)(i % HWPIX);
    int y = pix >> 7, x = pix & 127;
    float acc = 0.f;
    #pragma unroll
    for (int dy = -1; dy <= 1; ++dy) {
        #pragma unroll
        for (int dx = -1; dx <= 1; ++dx) {
            int yy = y + dy, xx = x + dx;
            if (yy >= 0 && yy < HH && xx >= 0 && xx < WW) {
                float w = (dy ? k0 : k1) * (dx ? k0 : k1);
                acc += w * map0[((size_t)b * HWPIX + yy * WW + xx) * 129 + 128];
            }
        }
    }
    bmask[i] = acc;
}

// ---------------------------------------------------------------------------
// Kernel 5: horizontal Gaussian pass as WMMA: Out[y] = T(tridiag) * In[y],
// M = output x (16), K = input x (6 fixed slices of 4, zero-pad masked),
// N = channels (16). One wave per (b, y, x-tile); loops all 8 channel tiles,
// reusing the analytic A-fragments. Fixed unroll -> uniform EXEC, batched
// loads, 6 back-to-back v_wmma per channel tile.
__global__ void k_hblur(const float* __restrict__ map0, float* __restrict__ tmp,
                        float k0, float k1) {
    int wid  = (int)(((size_t)blockIdx.x * blockDim.x + threadIdx.x) >> 5);
    int lane = threadIdx.x & 31;
    int mrow = lane & 15, grp = lane >> 4;
    int xt = wid & 7;
    int y  = (wid >> 3) & 127;
    int b  = wid >> 10;
    int m0 = xt << 4;
    int m  = m0 + mrow;
    const float* in   = map0 + (size_t)b * HWPIX * 129 + (size_t)y * WW * 129;
    float*       orow = tmp  + (size_t)b * HWPIX * CC  + (size_t)y * WW * CC;

    // Analytic A-fragments: weights (zero-pad masked) + clamped K addresses.
    float aw0[6], aw1[6];
    int   kc0[6], kc1[6];
    #pragma unroll
    for (int u = 0; u < 6; ++u) {
        int kg0 = 4 * (4 * xt - 1 + u) + 2 * grp;
        int kg1 = kg0 + 1;
        aw0[u] = (kg0 >= 0 && kg0 < WW) ? wtap(kg0 - m, k0, k1) : 0.f;
        aw1[u] = (kg1 >= 0 && kg1 < WW) ? wtap(kg1 - m, k0, k1) : 0.f;
        kc0[u] = kg0 < 0 ? 0 : (kg0 > WW - 1 ? WW - 1 : kg0);
        kc1[u] = kg1 < 0 ? 0 : (kg1 > WW - 1 ? WW - 1 : kg1);
    }
    for (int ct = 0; ct < 8; ++ct) {
        int cc = (ct << 4) + mrow;   // this lane's channel
        v8f acc = {};
        #pragma unroll
        for (int u = 0; u < 6; ++u) {
            v2f a, bf;
            a.x  = aw0[u];
            a.y  = aw1[u];
            bf.x = in[(size_t)kc0[u] * 129 + cc];
            bf.y = in[(size_t)kc1[u] * 129 + cc];
            acc = __builtin_amdgcn_wmma_f32_16x16x4_f32(
                false, a, false, bf, (short)0, acc, false, false);
        }
        #pragma unroll
        for (int r = 0; r < 8; ++r) {
            int xo = m0 + r + 8 * grp;
            orow[(size_t)xo * CC + cc] = acc[r];
        }
    }
}

// ---------------------------------------------------------------------------
// Kernel 6: vertical Gaussian pass as WMMA (M = output y, N = channels),
// fused with the reconstruction: out = f + (1-m) * gated(blur_f / blur_m).
// One wave per (b, x, y-tile); loops all 8 channel tiles; per-pixel mask and
// blurred-mask loads hoisted out of the channel loop.
__global__ void k_vblur(const float* __restrict__ tmp, const float* __restrict__ map0,
                        const float* __restrict__ bmask, float* __restrict__ map1,
                        float k0, float k1) {
    int wid  = (int)(((size_t)blockIdx.x * blockDim.x + threadIdx.x) >> 5);
    int lane = threadIdx.x & 31;
    int mrow = lane & 15, grp = lane >> 4;
    int yt = wid & 7;
    int x  = (wid >> 3) & 127;
    int b  = wid >> 10;
    int m0 = yt << 4;
    int m  = m0 + mrow;
    const float* in  = tmp   + (size_t)b * HWPIX * CC;
    const float* m0p = map0  + (size_t)b * HWPIX * 129;
    const float* bmp = bmask + (size_t)b * HWPIX;
    float*       o   = map1  + (size_t)b * HWPIX * CC;

    float aw0[6], aw1[6];
    int   kc0[6], kc1[6];
    #pragma unroll
    for (int u = 0; u < 6; ++u) {
        int kg0 = 4 * (4 * yt - 1 + u) + 2 * grp;
        int kg1 = kg0 + 1;
        aw0[u] = (kg0 >= 0 && kg0 < HH) ? wtap(kg0 - m, k0, k1) : 0.f;
        aw1[u] = (kg1 >= 0 && kg1 < HH) ? wtap(kg1 - m, k0, k1) : 0.f;
        kc0[u] = kg0 < 0 ? 0 : (kg0 > HH - 1 ? HH - 1 : kg0);
        kc1[u] = kg1 < 0 ? 0 : (kg1 > HH - 1 ? HH - 1 : kg1);
    }
    // Per-output-pixel mask state (same for all 8 channel tiles).
    float mskv[8], rgate[8];
    size_t pixv[8];
    #pragma unroll
    for (int r = 0; r < 8; ++r) {
        int yo = m0 + r + 8 * grp;
        size_t pix = (size_t)yo * WW + x;
        pixv[r]  = pix;
        mskv[r]  = m0p[pix * 129 + 128];
        float bm = bmp[pix];
        float g  = bm > 0.f ? 1.f : 0.f;
        rgate[r] = g / (bm + EPSV);        // gated reciprocal of blurred mask
    }
    for (int ct = 0; ct < 8; ++ct) {
        int cc = (ct << 4) + mrow;
        v8f acc = {};
        #pragma unroll
        for (int u = 0; u < 6; ++u) {
            v2f a, bf;
            a.x  = aw0[u];
            a.y  = aw1[u];
            bf.x = in[((size_t)kc0[u] * WW + x) * CC + cc];
            bf.y = in[((size_t)kc1[u] * WW + x) * CC + cc];
            acc = __builtin_amdgcn_wmma_f32_16x16x4_f32(
                false, a, false, bf, (short)0, acc, false, false);
        }
        #pragma unroll
        for (int r = 0; r < 8; ++r) {
            float f  = m0p[pixv[r] * 129 + cc];
            float fi = acc[r] * rgate[r];
            o[pixv[r] * CC + cc] = f + (1.f - mskv[r]) * fi;
        }
    }
}

// ---------------------------------------------------------------------------
// Kernel 7: bilinear gather. One wave per token; lane handles 4 channels
// (float4), 4 corner samples with zero-padding validity, coalesced output.
__global__ void k_gather(const float* __restrict__ map1, const float* __restrict__ loc,
                         float* __restrict__ out) {
    int token = blockIdx.x * 8 + (threadIdx.x >> 5);
    int lane = threadIdx.x & 31;
    int b = token >> 14;
    float lx = loc[(size_t)token * 2 + 0];
    float ly = loc[(size_t)token * 2 + 1];
    float px = lx * WW - 0.5f;
    float py = ly * HH - 0.5f;
    float fx0 = floorf(px), fy0 = floorf(py);
    float wx = px - fx0, wy = py - fy0;
    int x0 = (int)fx0, y0 = (int)fy0;
    const float* fm = map1 + (size_t)b * HWPIX * CC;
    int c = lane * 4;

    auto samp = [&](int yy, int xx) -> v4f {
        bool v = (xx >= 0) && (xx < WW) && (yy >= 0) && (yy < HH);
        int xc = xx < 0 ? 0 : (xx > WW - 1 ? WW - 1 : xx);
        int yc = yy < 0 ? 0 : (yy > HH - 1 ? HH - 1 : yy);
        v4f val = *(const v4f*)(fm + ((size_t)yc * WW + xc) * CC + c);
        return v ? val : (v4f){0.f, 0.f, 0.f, 0.f};
    };
    v4f v00 = samp(y0, x0);
    v4f v01 = samp(y0, x0 + 1);
    v4f v10 = samp(y0 + 1, x0);
    v4f v11 = samp(y0 + 1, x0 + 1);
    float w00 = (1.f - wx) * (1.f - wy), w01 = wx * (1.f - wy);
    float w10 = (1.f - wx) * wy,         w11 = wx * wy;
    v4f r = v00 * w00 + v01 * w01 + v10 * w10 + v11 * w11;
    *(v4f*)(out + (size_t)token * CC + c) = r;
}

// ---------------------------------------------------------------------------
extern "C" void kernel_launch(void* const* d_in, const int* in_sizes, int n_in,
                              void* d_out, int out_size, void* d_ws, size_t ws_size,
                              hipStream_t stream) {
    const float* x   = (const float*)d_in[0];
    const float* loc = (const float*)d_in[1];
    float* out = (float*)d_out;

    // Workspace layout (fp32):
    //   map0  : B*HW*129  (scatter accumulator -> normalized feature+mask)
    //   tmp   : B*HW*128  (horizontal blur intermediate)
    //   map1  : B*HW*128  (reconstructed feature map)
    //   bmask : B*HW      (blurred mask channel)
    float* map0  = (float*)d_ws;
    float* tmp   = map0 + (size_t)BB * HWPIX * 129;
    float* map1  = tmp  + (size_t)BB * HWPIX * CC;
    float* bmask = map1 + (size_t)BB * HWPIX * CC;

    // Separable normalized 1-D Gaussian taps: k = [a,1,a]/(2a+1), a=exp(-1/(2*sigma^2))
    double a = exp(-1.0 / 8.0);  // sigma = 2.0
    float k1v = (float)(1.0 / (2.0 * a + 1.0));  // center tap
    float k0v = (float)(a / (2.0 * a + 1.0));    // side tap

    k_zero     <<<33024,  256, 0, stream>>>(map0);            // B*HW*129/4 float4 stores
    k_scatter  <<<131072, 256, 0, stream>>>(x, loc, map0);    // 2 tokens / block
    k_norm_feat<<<131072, 256, 0, stream>>>(map0);            // B*HW*128 threads
    k_norm_mask<<<8192,   256, 0, stream>>>(map0);            // B*HW threads
    k_bmask    <<<8192,   256, 0, stream>>>(map0, bmask, k0v, k1v);
    k_hblur    <<<2048,   256, 0, stream>>>(map0, tmp, k0v, k1v);   // 16384 waves, 48 WMMA each
    k_vblur    <<<2048,   256, 0, stream>>>(tmp, map0, bmask, map1, k0v, k1v);
    k_gather   <<<32768,  256, 0, stream>>>(map1, loc, out);  // 8 tokens / block
}